// MedianBlur_91164975824930
// MI455X (gfx1250) — compile-verified
//
#include <hip/hip_runtime.h>

// ---------------------------------------------------------------------------
// MedianBlur 3x3, fp32, zero padding.  Input (8,3,1024,1024) -> same shape.
// Memory-bound (201 MB @ 23.3 TB/s ~ 8.6 us).  Strategy: TDM tile load to
// LDS (CDNA5 tensor_load_to_lds, iterate mode for LDS row stride + OOB
// zero-fill), then exact 3x3 median via column-sort3 / max-med-min network.
// ---------------------------------------------------------------------------

constexpr int IMG_H   = 1024;
constexpr int IMG_W   = 1024;
constexpr int NPLANES = 8 * 3;

constexpr int TW      = 256;      // output tile width  per block
constexpr int TH      = 16;       // output tile height per block
constexpr int LDSROWS = TH + 2;   // 18 rows incl. halo
constexpr int LDSW    = 260;      // padded row stride (floats): 16B-aligned rows

#if defined(__has_builtin)
#if __has_builtin(__builtin_amdgcn_tensor_load_to_lds)
#define HAS_TDM 1
#endif
#endif
#ifndef HAS_TDM
#define HAS_TDM 0
#endif

#if HAS_TDM
typedef unsigned int u32x4 __attribute__((ext_vector_type(4)));
typedef int          i32x4 __attribute__((ext_vector_type(4)));
typedef int          i32x8 __attribute__((ext_vector_type(8)));
#endif

__device__ __forceinline__ float med3f(float a, float b, float c) {
  // median of 3: max(min(a,b), min(max(a,b), c))
  return fmaxf(fminf(a, b), fminf(fmaxf(a, b), c));
}

__device__ __forceinline__ void load6(const float* p, float v[6]) {
  // p is 16B aligned (row stride 260 floats -> 1040B, col base 16*tx bytes)
  const float4 a = *(const float4*)(p);
  const float2 b = *(const float2*)(p + 4);
  v[0] = a.x; v[1] = a.y; v[2] = a.z; v[3] = a.w; v[4] = b.x; v[5] = b.y;
}

__global__ __launch_bounds__(256) void median3x3_tdm(const float* __restrict__ in,
                                                     float* __restrict__ out) {
  __shared__ __align__(16) float lds[LDSROWS * LDSW];

  const int  tid       = threadIdx.x;
  const int  x0        = blockIdx.x * TW;
  const int  y0        = blockIdx.y * TH;
  const long planeBase = (long)blockIdx.z * (long)(IMG_H * IMG_W);

#if HAS_TDM
  // ---- stage halo tile into LDS with the Tensor Data Mover ----------------
  // Pre-zero LDS only for tiles touching a zero-padded border (left/top need
  // a shifted destination; bottom row is not loaded). Right edge is zero-
  // filled by TDM OOB handling (tensor_dim0 < tile_dim0).
  if ((x0 == 0) | (y0 == 0) | (y0 + TH >= IMG_H)) {
    for (int i = tid; i < LDSROWS * LDSW; i += 256) lds[i] = 0.0f;
  }
  __syncthreads();  // LDS zeroing visible before DMA writes

  if (tid < 32) {   // wave 0 issues the DMA and waits on TENSORcnt
    const int srow = (y0 > 0) ? (y0 - 1) : 0;
    const int scol = (x0 > 0) ? (x0 - 1) : 0;
    const int erow = (y0 + TH < IMG_H) ? (y0 + TH) : (IMG_H - 1);
    const int lh   = erow - srow + 1;                  // rows actually loaded
    const int lw   = TW + 2 - ((x0 == 0) ? 1 : 0);     // tile_dim0 (elements)

    // LDS byte address of the first loaded element (shift for clamped edges).
    const unsigned ldsOff =
        (unsigned)(uintptr_t)(&lds[0]) +
        4u * (unsigned)(((y0 == 0) ? LDSW : 0) + ((x0 == 0) ? 1 : 0));

    const unsigned long long ga =
        (unsigned long long)(uintptr_t)(in + planeBase + (long)srow * IMG_W + scol);

    const unsigned td0 = (unsigned)(IMG_W - scol);     // remaining width  (OOB->0)
    const unsigned td1 = (unsigned)(IMG_H - srow);     // remaining height

    // D# group 0: count=1 | lds_addr | global_addr[56:0] | type=2
    u32x4 g0 = { 1u,
                 ldsOff,
                 (unsigned)(ga & 0xFFFFFFFFull),
                 (unsigned)((ga >> 32) & 0x01FFFFFFull) | 0x80000000u };
    // D# group 1: data_size=4B (2), iterate_enable=1; dims/strides per layout
    i32x8 g1 = { (int)0x000A0000u,                    // data_size=2, iterate=1
                 (int)(td0 << 16),                    // tensor_dim0[15:0]
                 (int)(td1 << 16),                    // td0 hi=0 | tensor_dim1 lo
                 (int)((unsigned)lw << 16),           // td1 hi=0 | tile_dim0
                 1,                                   // tile_dim1=1, tile_dim2=0
                 IMG_W,                               // tensor_dim0_stride lo
                 (int)((unsigned)IMG_W << 16),        // stride0 hi=0 | stride1 lo
                 0 };                                 // stride1 hi
    // D# group 2 (iterate mode): lds_addr_inc=LDSW, global_addr_inc=W,
    // iterate_count = lh-1  (0 => 1 row)
    i32x4 g2 = { 0, LDSW, IMG_W, (int)((unsigned)(lh - 1) << 16) };
    i32x4 g3 = { 0, 0, 0, 0 };

#if __has_include(<hip/amd_detail/amd_gfx1250_TDM.h>)
    i32x8 g4 = { 0, 0, 0, 0, 0, 0, 0, 0 };
    __builtin_amdgcn_tensor_load_to_lds(g0, g1, g2, g3, g4, 0);
#else
    __builtin_amdgcn_tensor_load_to_lds(g0, g1, g2, g3, 0);
#endif
#if __has_builtin(__builtin_amdgcn_s_wait_tensorcnt)
    __builtin_amdgcn_s_wait_tensorcnt(0);
#else
    asm volatile("s_wait_tensorcnt 0x0" ::: "memory");
#endif
  }
#else
  // ---- fallback: cooperative vector-lane loads with bounds checks ---------
  for (int i = tid; i < LDSROWS * (TW + 2); i += 256) {
    const int r  = i / (TW + 2);
    const int c  = i % (TW + 2);
    const int gy = y0 - 1 + r;
    const int gx = x0 - 1 + c;
    float v = 0.0f;
    if (gy >= 0 && gy < IMG_H && gx >= 0 && gx < IMG_W)
      v = in[planeBase + (long)gy * IMG_W + gx];
    lds[r * LDSW + c] = v;
  }
#endif
  __syncthreads();

  // ---- compute: each thread does a 4 wide x 4 tall patch ------------------
  const int tx = tid & 63;        // 64 threads across -> 256 px
  const int ty = tid >> 6;        // 4 threads down -> 4 px each
  const int cb = tx * 4;          // LDS column of leftmost window column
  const int rb = ty * 4;          // LDS row of topmost window row

  float r0[6], r1[6], r2[6];      // 3 sliding image rows x 6 columns
  load6(&lds[(rb + 0) * LDSW + cb], r0);
  load6(&lds[(rb + 1) * LDSW + cb], r1);
  load6(&lds[(rb + 2) * LDSW + cb], r2);

#pragma unroll
  for (int k = 0; k < 4; ++k) {
    // sort each of the 6 columns (lo/mi/hi), shared by 4 horizontal outputs
    float lo[6], mi[6], hi[6];
#pragma unroll
    for (int j = 0; j < 6; ++j) {
      const float t0 = fminf(r0[j], r1[j]);
      const float t1 = fmaxf(r0[j], r1[j]);
      lo[j] = fminf(t0, r2[j]);
      hi[j] = fmaxf(t1, r2[j]);
      mi[j] = fmaxf(t0, fminf(t1, r2[j]));
    }
    float o[4];
#pragma unroll
    for (int j = 0; j < 4; ++j) {
      const float mx = fmaxf(fmaxf(lo[j], lo[j + 1]), lo[j + 2]);
      const float mn = fminf(fminf(hi[j], hi[j + 1]), hi[j + 2]);
      const float md = med3f(mi[j], mi[j + 1], mi[j + 2]);
      o[j] = med3f(mx, md, mn);   // exact 3x3 median
    }
    float4* dst = (float4*)(out + planeBase + (long)(y0 + rb + k) * IMG_W + (x0 + cb));
    *dst = make_float4(o[0], o[1], o[2], o[3]);

    if (k < 3) {                  // slide window down one row
#pragma unroll
      for (int j = 0; j < 6; ++j) { r0[j] = r1[j]; r1[j] = r2[j]; }
      load6(&lds[(rb + 3 + k) * LDSW + cb], r2);
    }
  }
}

extern "C" void kernel_launch(void* const* d_in, const int* in_sizes, int n_in,
                              void* d_out, int out_size, void* d_ws, size_t ws_size,
                              hipStream_t stream) {
  (void)in_sizes; (void)n_in; (void)d_ws; (void)ws_size; (void)out_size;
  const float* in  = (const float*)d_in[0];
  float*       out = (float*)d_out;
  dim3 grid(IMG_W / TW, IMG_H / TH, NPLANES);   // (4, 64, 24)
  median3x3_tdm<<<grid, dim3(256), 0, stream>>>(in, out);
}